// E3DensityModel_3315714752865
// MI455X (gfx1250) — compile-verified
//
#include <hip/hip_runtime.h>
#include <hip/hip_bf16.h>
#include <math.h>

// ---------------------------------------------------------------------------
// E3 equivariant conv, fused for MI455X (gfx1250, wave32, WMMA).
//   K0: zero agg buffers
//   K1: node projections f_s/f_v/sc_s/sc_v (VALU, tiny)
//   K2: fused edge MLP (V_WMMA_F32_16X16X4_F32) + tensor product + atomic
//       scatter into agg_s/agg_v  (the 18.7 GFLOP + 192M-atomic hot loop)
//   K3: node output GEMVs + cos/sin mixing
// ---------------------------------------------------------------------------

typedef __attribute__((ext_vector_type(2))) float v2f;
typedef __attribute__((ext_vector_type(8))) float v8f;

#define MUL0   64
#define MUL1   32
#define NBASIS 10
#define FCH    100
#define WNUM   224   // 2*MUL0 + 3*MUL1
#define MIDS   96
#define MIDV   128
#define KPAD1  12    // NBASIS padded to multiple of 4
#define NPAD1  112   // FCH padded to multiple of 16
#define WAVES  4     // waves per block in edge kernel

// hardware f32 atomic add, no return (global_atomic_add_f32)
__device__ __forceinline__ void atomic_add_f32(float* p, float v) {
  __hip_atomic_fetch_add(p, v, __ATOMIC_RELAXED, __HIP_MEMORY_SCOPE_AGENT);
}

// fast reciprocal (v_rcp_f32), ~1 ulp — fine for silu
__device__ __forceinline__ float fast_rcp(float x) {
#if __has_builtin(__builtin_amdgcn_rcpf)
  return __builtin_amdgcn_rcpf(x);
#else
  return 1.f / x;
#endif
}

// -------------------------------- K0 ---------------------------------------
__global__ void zero_kernel(float* __restrict__ p, long long n) {
  long long i = (long long)blockIdx.x * blockDim.x + threadIdx.x;
  long long stride = (long long)gridDim.x * blockDim.x;
  for (; i < n; i += stride) p[i] = 0.f;
}

// -------------------------------- K1 ---------------------------------------
// block = 160 threads (5 waves), one node per block.
__global__ void node_prep_kernel(const float* __restrict__ ns, const float* __restrict__ nv,
                                 const float* __restrict__ attr,
                                 const float* __restrict__ Wls, const float* __restrict__ Wlv,
                                 const float* __restrict__ Wss, const float* __restrict__ Wsv,
                                 float* __restrict__ f_s, float* __restrict__ f_v,
                                 float* __restrict__ sc_s, float* __restrict__ sc_v, int n)
{
  int node = blockIdx.x;
  if (node >= n) return;
  __shared__ float s_ns[MUL0];
  __shared__ float s_nv[MUL1 * 3];
  int t = threadIdx.x;
  if (t < MUL0)        s_ns[t]        = ns[(long long)node * MUL0 + t];
  else                 s_nv[t - MUL0] = nv[(long long)node * (MUL1 * 3) + (t - MUL0)];
  __syncthreads();
  float a = attr[node];
  if (t < MUL0) {
    float af = 0.f, asc = 0.f;
    #pragma unroll 8
    for (int u = 0; u < MUL0; ++u) {
      float x = s_ns[u];
      af  += x * Wls[u * MUL0 + t];
      asc += x * Wss[u * MUL0 + t];
    }
    f_s [(long long)node * MUL0 + t] = af  * a;
    sc_s[(long long)node * MUL0 + t] = asc * a;
  } else {
    int idx = t - MUL0;            // 0..95 -> (v, c)
    int vv = idx / 3;
    int k  = idx - vv * 3;
    float af = 0.f, asc = 0.f;
    #pragma unroll 8
    for (int u = 0; u < MUL1; ++u) {
      float x = s_nv[u * 3 + k];
      af  += x * Wlv[u * MUL1 + vv];
      asc += x * Wsv[u * MUL1 + vv];
    }
    f_v [(long long)node * (MUL1 * 3) + idx] = af  * a;
    sc_v[(long long)node * (MUL1 * 3) + idx] = asc * a;
  }
}

// -------------------------------- K2 ---------------------------------------
// One wave <-> one tile of 16 edges.
//  GEMM1: (16 x 12) @ (12 x 112)  -> 7 ntiles x 3 ksteps  = 21 WMMAs
//  GEMM2: (16 x 112) @ (112 x 224)-> 14 ntiles x 28 ksteps = 392 WMMAs
//  Then the tensor product consumes the C-layout accumulator directly:
//  acc[r] @ lane L holds w[edge m = r + 8*(L>=16)][chan c = 16*nt + (L&15)].
__global__ __launch_bounds__(WAVES * 32)
void edge_fused_kernel(const float* __restrict__ es,   // edge_scalars (E,10)
                       const float* __restrict__ ea,   // edge_attr    (E,4)
                       const int*   __restrict__ esrc,
                       const int*   __restrict__ edst,
                       const float* __restrict__ Wfc1, // (10,100)
                       const float* __restrict__ Wfc2, // (100,224)
                       const float* __restrict__ f_s,  // (N,64)
                       const float* __restrict__ f_v,  // (N,32,3)
                       float* __restrict__ agg_s,      // (N,96)
                       float* __restrict__ agg_v,      // (N,128,3)
                       int E, int numTiles)
{
  __shared__ float W1s[KPAD1 * NPAD1];        //   5.25 KB, zero-padded W_fc1
  __shared__ float W2s[NPAD1 * WNUM];         // 100.4 KB, zero-padded W_fc2
  __shared__ float Hs[WAVES][16 * NPAD1];     //  28 KB, per-wave hidden tile
  __shared__ float mF[WAVES][16 * 4];         // per-wave edge attr (y0,y1xyz)
  __shared__ int   mI[WAVES][16 * 2];         // per-wave edge src/dst

  // ---- stage padded weights (whole block) ----
  for (int i = threadIdx.x; i < KPAD1 * NPAD1; i += blockDim.x) {
    int k = i / NPAD1, ncol = i - k * NPAD1;
    W1s[i] = (k < NBASIS && ncol < FCH) ? Wfc1[k * FCH + ncol] : 0.f;
  }
  for (int i = threadIdx.x; i < NPAD1 * WNUM; i += blockDim.x) {
    int k = i / WNUM;
    W2s[i] = (k < FCH) ? Wfc2[i] : 0.f;       // i == k*224 + n
  }
  __syncthreads();

  const int wid  = threadIdx.x >> 5;
  const int lane = threadIdx.x & 31;
  const int hi   = lane >> 4;    // 0: lanes 0-15, 1: lanes 16-31
  const int lm   = lane & 15;

  float* H  = Hs[wid];
  float* mf = mF[wid];
  int*   mi = mI[wid];

  const float inv  = 0.25f;                    // 1/sqrt(NUM_NEIGHBORS)
  const float inv3 = 0.57735026918962576451f;  // 1/sqrt(3)
  const float inv2 = 0.70710678118654752440f;  // 1/sqrt(2)

  int tile       = blockIdx.x * WAVES + wid;
  int tileStride = gridDim.x * WAVES;

  for (; tile < numTiles; tile += tileStride) {
    int base = tile * 16;

    // ---- per-tile edge metadata (lanes 0..15) ----
    if (lane < 16) {
      int e = base + lane;
      if (e < E) {
        mi[lane * 2 + 0] = esrc[e];
        mi[lane * 2 + 1] = edst[e];
        mf[lane * 4 + 0] = ea[(long long)e * 4 + 0];
        mf[lane * 4 + 1] = ea[(long long)e * 4 + 1];
        mf[lane * 4 + 2] = ea[(long long)e * 4 + 2];
        mf[lane * 4 + 3] = ea[(long long)e * 4 + 3];
      } else {
        mi[lane * 2 + 0] = 0;
        mi[lane * 2 + 1] = -1;                 // marks invalid edge
        mf[lane * 4 + 0] = 0.f; mf[lane * 4 + 1] = 0.f;
        mf[lane * 4 + 2] = 0.f; mf[lane * 4 + 3] = 0.f;
      }
    }

    // ---- A fragments for GEMM1: row m = lm, k0 = 4*kk + 2*hi ----
    int  eA = base + lm;
    bool ev = (eA < E);
    v2f a1[3];
    #pragma unroll
    for (int kk = 0; kk < 3; ++kk) {
      int k0 = 4 * kk + 2 * hi;
      a1[kk].x = (ev && (k0     < NBASIS)) ? es[(long long)eA * NBASIS + k0    ] : 0.f;
      a1[kk].y = (ev && (k0 + 1 < NBASIS)) ? es[(long long)eA * NBASIS + k0 + 1] : 0.f;
    }

    // ---- GEMM1 + silu -> H (per-wave LDS) ----
    #pragma unroll
    for (int nt = 0; nt < NPAD1 / 16; ++nt) {
      v8f acc = {};
      #pragma unroll
      for (int kk = 0; kk < 3; ++kk) {
        int k0 = 4 * kk + 2 * hi;
        v2f b;
        b.x = W1s[ k0      * NPAD1 + nt * 16 + lm];
        b.y = W1s[(k0 + 1) * NPAD1 + nt * 16 + lm];
        acc = __builtin_amdgcn_wmma_f32_16x16x4_f32(
            false, a1[kk], false, b, (short)0, acc, false, false);
      }
      #pragma unroll
      for (int r = 0; r < 8; ++r) {            // C layout: m = r + 8*hi
        float x  = acc[r];
        float sg = fast_rcp(1.f + __expf(-x)); // silu via v_exp + v_rcp
        H[(r + 8 * hi) * NPAD1 + nt * 16 + lm] = x * sg;
      }
    }
    // same-wave LDS RAW: wait stores, block compiler reordering
    asm volatile("s_wait_dscnt 0" ::: "memory");

    // ---- preload GEMM2 A fragments from H (row m = lm) ----
    v2f a2[NPAD1 / 4];
    #pragma unroll
    for (int kk = 0; kk < NPAD1 / 4; ++kk) {
      int k0 = 4 * kk + 2 * hi;                // even -> 8B aligned
      a2[kk] = *(const v2f*)&H[lm * NPAD1 + k0];
    }

    // ---- GEMM2 per ntile, then tensor product + scatter ----
    for (int nt = 0; nt < WNUM / 16; ++nt) {
      v8f acc = {};
      #pragma unroll
      for (int kk = 0; kk < NPAD1 / 4; ++kk) {
        int k0 = 4 * kk + 2 * hi;
        v2f b;
        b.x = W2s[ k0      * WNUM + nt * 16 + lm];
        b.y = W2s[(k0 + 1) * WNUM + nt * 16 + lm];
        acc = __builtin_amdgcn_wmma_f32_16x16x4_f32(
            false, a2[kk], false, b, (short)0, acc, false, false);
      }

      int c = nt * 16 + lm;                    // weight channel (uniform range per nt)
      #pragma unroll
      for (int r = 0; r < 8; ++r) {
        int m   = r + 8 * hi;                  // edge within tile
        int dst = mi[m * 2 + 1];
        if (dst < 0) continue;
        int   src = mi[m * 2 + 0];
        float wv  = acc[r];
        float y0  = mf[m * 4 + 0];
        float y1x = mf[m * 4 + 1], y1y = mf[m * 4 + 2], y1z = mf[m * 4 + 3];

        if (c < 64) {                          // w_00: mid_s[c] = w*s*y0
          float s = f_s[(long long)src * MUL0 + c];
          atomic_add_f32(&agg_s[(long long)dst * MIDS + c], wv * s * y0 * inv);
        } else if (c < 128) {                  // w_01: mid_v[u] = w*s*y1
          int   u = c - 64;
          float s = f_s[(long long)src * MUL0 + u];
          float t = wv * s * inv;
          atomic_add_f32(&agg_v[((long long)dst * MIDV + u) * 3 + 0], t * y1x);
          atomic_add_f32(&agg_v[((long long)dst * MIDV + u) * 3 + 1], t * y1y);
          atomic_add_f32(&agg_v[((long long)dst * MIDV + u) * 3 + 2], t * y1z);
        } else if (c < 160) {                  // w_10: mid_v[64+u] = w*v*y0
          int u = c - 128;
          const float* vp = &f_v[((long long)src * MUL1 + u) * 3];
          float t = wv * y0 * inv;
          atomic_add_f32(&agg_v[((long long)dst * MIDV + 64 + u) * 3 + 0], t * vp[0]);
          atomic_add_f32(&agg_v[((long long)dst * MIDV + 64 + u) * 3 + 1], t * vp[1]);
          atomic_add_f32(&agg_v[((long long)dst * MIDV + 64 + u) * 3 + 2], t * vp[2]);
        } else if (c < 192) {                  // w_11s: mid_s[64+u] = w*dot(v,y1)/sqrt3
          int u = c - 160;
          const float* vp = &f_v[((long long)src * MUL1 + u) * 3];
          float d = vp[0] * y1x + vp[1] * y1y + vp[2] * y1z;
          atomic_add_f32(&agg_s[(long long)dst * MIDS + 64 + u], wv * d * inv3 * inv);
        } else {                               // w_11v: mid_v[96+u] = w*cross(v,y1)/sqrt2
          int u = c - 192;
          const float* vp = &f_v[((long long)src * MUL1 + u) * 3];
          float cx = vp[1] * y1z - vp[2] * y1y;
          float cy = vp[2] * y1x - vp[0] * y1z;
          float cz = vp[0] * y1y - vp[1] * y1x;
          float t = wv * inv2 * inv;
          atomic_add_f32(&agg_v[((long long)dst * MIDV + 96 + u) * 3 + 0], t * cx);
          atomic_add_f32(&agg_v[((long long)dst * MIDV + 96 + u) * 3 + 1], t * cy);
          atomic_add_f32(&agg_v[((long long)dst * MIDV + 96 + u) * 3 + 2], t * cz);
        }
      }
    }
  }
}

// -------------------------------- K3 ---------------------------------------
// block = 160 threads, one node per block.
__global__ void node_out_kernel(const float* __restrict__ agg_s, const float* __restrict__ agg_v,
                                const float* __restrict__ sc_s,  const float* __restrict__ sc_v,
                                const float* __restrict__ attr,
                                const float* __restrict__ Wl2s, const float* __restrict__ Wl2v,
                                const float* __restrict__ Wl3,
                                float* __restrict__ out, int n)
{
  int node = blockIdx.x;
  if (node >= n) return;
  __shared__ float sAs[MIDS];
  __shared__ float sAv[MIDV * 3];
  int t = threadIdx.x;
  if (t < MIDS) sAs[t] = agg_s[(long long)node * MIDS + t];
  for (int i = t; i < MIDV * 3; i += 160) sAv[i] = agg_v[(long long)node * (MIDV * 3) + i];
  __syncthreads();

  float a = attr[node];
  float ang = 0.f;
  #pragma unroll 8
  for (int u = 0; u < MIDS; ++u) ang += sAs[u] * Wl3[u];
  ang *= 0.1f * a;
  float cs = cosf(ang);
  float sn = sinf(ang);

  if (t < MUL0) {
    float conv = 0.f;
    #pragma unroll 8
    for (int u = 0; u < MIDS; ++u) conv += sAs[u] * Wl2s[u * MUL0 + t];
    conv *= a;
    out[(long long)node * 160 + t] = cs * sc_s[(long long)node * MUL0 + t] + sn * conv;
  } else {
    int idx = t - MUL0;
    int vv = idx / 3;
    int k  = idx - vv * 3;
    float conv = 0.f;
    #pragma unroll 8
    for (int u = 0; u < MIDV; ++u) conv += sAv[u * 3 + k] * Wl2v[u * MUL1 + vv];
    conv *= a;
    out[(long long)node * 160 + t] = cs * sc_v[(long long)node * (MUL1 * 3) + idx] + sn * conv;
  }
}

// ---------------------------------------------------------------------------
extern "C" void kernel_launch(void* const* d_in, const int* in_sizes, int n_in,
                              void* d_out, int out_size, void* d_ws, size_t ws_size,
                              hipStream_t stream) {
  const float* node_scalars = (const float*)d_in[0];
  const float* node_vectors = (const float*)d_in[1];
  const float* node_attr    = (const float*)d_in[2];
  const int*   edge_src     = (const int*)  d_in[3];
  const int*   edge_dst     = (const int*)  d_in[4];
  const float* edge_attr    = (const float*)d_in[5];
  const float* edge_scalars = (const float*)d_in[6];
  const float* W_fc1        = (const float*)d_in[7];
  const float* W_fc2        = (const float*)d_in[8];
  const float* W_sc_s       = (const float*)d_in[9];
  const float* W_sc_v       = (const float*)d_in[10];
  const float* W_l1_s       = (const float*)d_in[11];
  const float* W_l1_v       = (const float*)d_in[12];
  const float* W_l2_s       = (const float*)d_in[13];
  const float* W_l2_v       = (const float*)d_in[14];
  const float* W_l3         = (const float*)d_in[15];

  const int N = in_sizes[0] / MUL0;
  const int E = in_sizes[3];

  // workspace layout (floats): f_s | f_v | sc_s | sc_v | agg_s | agg_v
  float* ws    = (float*)d_ws;
  float* f_s   = ws;
  float* f_v   = f_s  + (size_t)N * MUL0;
  float* sc_s  = f_v  + (size_t)N * MUL1 * 3;
  float* sc_v  = sc_s + (size_t)N * MUL0;
  float* agg_s = sc_v + (size_t)N * MUL1 * 3;
  float* agg_v = agg_s + (size_t)N * MIDS;

  // K0: zero agg_s+agg_v (contiguous)
  long long nAgg = (long long)N * (MIDS + MIDV * 3);
  zero_kernel<<<4096, 256, 0, stream>>>(agg_s, nAgg);

  // K1: node projections
  node_prep_kernel<<<N, 160, 0, stream>>>(node_scalars, node_vectors, node_attr,
                                          W_l1_s, W_l1_v, W_sc_s, W_sc_v,
                                          f_s, f_v, sc_s, sc_v, N);

  // K2: fused edge MLP (WMMA) + tensor product + scatter
  int numTiles = (E + 15) / 16;
  edge_fused_kernel<<<1024, WAVES * 32, 0, stream>>>(edge_scalars, edge_attr,
                                                     edge_src, edge_dst,
                                                     W_fc1, W_fc2, f_s, f_v,
                                                     agg_s, agg_v, E, numTiles);

  // K3: node output
  node_out_kernel<<<N, 160, 0, stream>>>(agg_s, agg_v, sc_s, sc_v, node_attr,
                                         W_l2_s, W_l2_v, W_l3, (float*)d_out, N);
}